// Decoder_15539191677793
// MI455X (gfx1250) — compile-verified
//
#include <hip/hip_runtime.h>

// Problem constants from the reference
#define NXC 512
#define NYC 512
#define PHW 3
#define PP 6            // patch size = 2*PATCH_HW
#define SPTS 1024       // points per image
#define BAND_ROWS 32    // rows per LDS band -> 32*512*4 = 64 KB LDS
#define BANDS (NXC / BAND_ROWS)   // 16
#define TPB 256         // 8 wave32s

typedef unsigned int uint32;
typedef uint32 v4u __attribute__((ext_vector_type(4)));
typedef int    v8i __attribute__((ext_vector_type(8)));
typedef int    v4i __attribute__((ext_vector_type(4)));

__global__ __launch_bounds__(TPB)
void psf_scatter_band(const float* __restrict__ z, float* __restrict__ out)
{
    __shared__ float lds[BAND_ROWS * NYC];   // 64 KB accumulation band

    const int band = blockIdx.x;             // 0..15  (row band within image)
    const int b    = blockIdx.y;             // 0..B-1 (image)
    const int tid  = threadIdx.x;
    const int r0   = band * BAND_ROWS;       // first image row of this band

    // ---- zero the LDS band (ds_store_b128, fully covers band) ----
    float4* l4 = (float4*)lds;
    #pragma unroll
    for (int i = 0; i < (BAND_ROWS * NYC / 4) / TPB; ++i)
        l4[tid + i * TPB] = make_float4(0.f, 0.f, 0.f, 0.f);
    __syncthreads();

    // constants: alpha = sqrt(2)*0.92 ; i0 = eta*N0*texp = 1000
    const float inva = 0.76859334f;          // 1/alpha
    const float i0   = 1000.0f;

    const float* zb = z + (size_t)b * (2 * SPTS);

    // ---- scan all points of this image; bin patches overlapping this band ----
    for (int p = tid; p < SPTS; p += TPB) {
        float x0 = zb[p];
        float y0 = zb[SPTS + p];
        int px = (int)rintf(x0) - PHW;       // patch top row
        int py = (int)rintf(y0) - PHW;       // patch left col
        bool valid = (px >= 0) & (px < NXC - PP) & (py >= 0) & (py < NYC - PP);
        // row-range overlap with band [r0, r0+BAND_ROWS)
        if (!valid || px > r0 + BAND_ROWS - 1 || px + (PP - 1) < r0) continue;

        float x0p = x0 - (float)px;
        float y0p = y0 - (float)py;

        float fx[PP + 1], fy[PP + 1];
        #pragma unroll
        for (int k = 0; k <= PP; ++k) {
            float t = (float)k - 0.5f;
            fx[k] = erff((t - x0p) * inva);
            fy[k] = erff((t - y0p) * inva);
        }
        float wx[PP], wy[PP];
        #pragma unroll
        for (int k = 0; k < PP; ++k) {
            wx[k] = 0.5f * i0 * (fx[k + 1] - fx[k]);   // fold i0 into row weights
            wy[k] = 0.5f * (fy[k + 1] - fy[k]);
        }
        #pragma unroll
        for (int i = 0; i < PP; ++i) {
            int r = px + i;
            if (r < r0 || r >= r0 + BAND_ROWS) continue;
            int base = (r - r0) * NYC + py;
            #pragma unroll
            for (int j = 0; j < PP; ++j)
                atomicAdd(&lds[base + j], wx[i] * wy[j]);   // ds_add_f32
        }
    }
    __syncthreads();

    // ---- write the band out: one TDM bulk store LDS -> global ----
    float* dst = out + ((size_t)b * NXC + (size_t)r0) * NYC;   // uniform per WG

#if __has_builtin(__builtin_amdgcn_tensor_store_from_lds)
    if (tid < 32) {                       // one wave issues the tensor DMA
        unsigned long long ga = (unsigned long long)(uintptr_t)dst;
        uint32 lo = (uint32)(uintptr_t)(void*)&lds[0];   // low 32 bits = LDS offset

        // D# group 0: count=1 | lds_addr | global_addr(57b) | type=2
        v4u g0;
        g0.x = 1u;                                        // count = 1 (valid)
        g0.y = lo;                                        // lds_addr
        g0.z = (uint32)(ga & 0xFFFFFFFFull);              // global_addr[31:0]
        g0.w = (uint32)((ga >> 32) & 0x01FFFFFFull)       // global_addr[56:32]
             | (2u << 30);                                // type = 2 ("image")

        // D# group 1: 1-D tile of BAND_ROWS*NYC fp32 elements
        const uint32 n = BAND_ROWS * NYC;                 // 16384 elements
        v8i g1;
        g1[0] = (int)(2u << 16);                          // data_size = 4 bytes
        g1[1] = (int)(n << 16);                           // tensor_dim0[15:0]
        g1[2] = (int)(1u << 16);                          // tensor_dim1 = 1
        g1[3] = (int)(n << 16);                           // tile_dim0 = n
        g1[4] = 0;                                        // tile_dim1/2 unused
        g1[5] = (int)n;                                   // tensor_dim0_stride
        g1[6] = 0;
        g1[7] = 0;

        v4i g2 = {0, 0, 0, 0};                            // group 2 (disabled)
        v4i g3 = {0, 0, 0, 0};                            // group 3 (disabled)
        v8i g4 = {0, 0, 0, 0, 0, 0, 0, 0};                // extra group (clang-23 form)

        // amdgpu-toolchain (clang-23) 6-arg form:
        // (uint32x4 g0, int32x8 g1, int32x4, int32x4, int32x8, i32 cpol)
        __builtin_amdgcn_tensor_store_from_lds(g0, g1, g2, g3, g4, 0);
        __builtin_amdgcn_s_wait_tensorcnt(0);             // s_wait_tensorcnt 0
    }
#else
    // Fallback: coalesced float4 stores
    const float4* s4 = (const float4*)lds;
    float4* d4 = (float4*)dst;
    for (int i = tid; i < BAND_ROWS * NYC / 4; i += TPB)
        d4[i] = s4[i];
#endif
}

extern "C" void kernel_launch(void* const* d_in, const int* in_sizes, int n_in,
                              void* d_out, int out_size, void* d_ws, size_t ws_size,
                              hipStream_t stream)
{
    const float* z = (const float*)d_in[0];
    float* out = (float*)d_out;
    int B = in_sizes[0] / (2 * SPTS);        // 128
    dim3 grid(BANDS, B);                     // 16 x 128 = 2048 workgroups
    psf_scatter_band<<<grid, TPB, 0, stream>>>(z, out);
}